// PEPS_Model_78941498901192
// MI455X (gfx1250) — compile-verified
//
#include <hip/hip_runtime.h>
#include <hip/hip_bf16.h>

// PEPS amplitude via boundary-MPS contraction, one workgroup (8 wave32s) per
// batch configuration. All tensor contractions are mapped to fp32 WMMA
// (V_WMMA_F32_16X16X4_F32). SVD truncation is replaced by a matmul-only
// Gram + Newton-Schulz subspace projection (equivalent row-space keep).

typedef __attribute__((ext_vector_type(2))) float v2f;
typedef __attribute__((ext_vector_type(8))) float v8f;

#if defined(__has_builtin)
#if __has_builtin(__builtin_amdgcn_wmma_f32_16x16x4_f32)
#define HAVE_WMMA_F32X4 1
#endif
#endif

namespace {

constexpr int LX = 10, LY = 10, PHYS = 2, D = 4, CHI = 32, BATCH = 64;
constexpr int NT = 256;            // threads per block
constexpr int WAVES = NT / 32;     // 8 wave32s
constexpr int BD  = CHI;           // padded bond dim (32)
constexpr int BDX = BD * D;        // expanded bond after absorb (128)

// per-block workspace layout (floats)
constexpr int AS_SZ  = LX * LY * 256;        // gathered site tensors (25600)
constexpr int MPS_SZ = LX * BD * D * BD;     // boundary MPS, [i][a][p][b] (40960)
constexpr int XS_SZ  = BDX * D * BDX;        // one expanded site (65536)
constexpr int X_SZ   = LX * XS_SZ;           // all expanded sites (655360)
constexpr int SC_SZ  = XS_SZ;                // GEMM scratch (65536)
constexpr int G_SZ   = BDX * BDX;            // Gram matrix (16384)

constexpr int OFF_AS = 0;
constexpr int OFF_L  = OFF_AS + AS_SZ;
constexpr int OFF_R  = OFF_L + MPS_SZ;
constexpr int OFF_X  = OFF_R + MPS_SZ;
constexpr int OFF_SC = OFF_X + X_SZ;
constexpr int OFF_G  = OFF_SC + SC_SZ;
constexpr int WS_PER = OFF_G + G_SZ;         // 844800 floats per block

__device__ inline void zero_f(float* p, int n) {
  for (int i = threadIdx.x; i < n; i += NT) p[i] = 0.f;
}
__device__ inline void copy_f(float* d, const float* s, int n) {
  for (int i = threadIdx.x; i < n; i += NT) d[i] = s[i];
}

// Block-cooperative GEMM: C[M,N] = op(A)[M,K] * op(B)[K,N], row-major.
// TA: A accessed transposed (A is K x M). TB: B accessed transposed (N x K).
// M,N multiples of 16; K multiple of 4. Each wave owns 16x16 output tiles and
// runs a K-loop of V_WMMA_F32_16X16X4_F32. EXEC is all-ones for every WMMA.
// The tile index is wave-uniform and passed through readfirstlane so the tile
// loop compiles as a scalar (SGPR-counter) loop instead of an exec-masked
// vector loop. Contiguous fragment pairs are fetched as single b64 loads.
template <bool TA, bool TB>
__device__ inline void gemm(const float* __restrict__ A, int lda,
                            const float* __restrict__ B, int ldb,
                            float* __restrict__ C, int ldc,
                            int M, int N, int K) {
  const int lane = threadIdx.x & 31;
  const int wave = __builtin_amdgcn_readfirstlane(threadIdx.x >> 5);  // uniform
  const int half = lane >> 4;   // 0 or 1
  const int l16  = lane & 15;
  const int mT = M >> 4, nT = N >> 4;
  const int tiles = mT * nT;
  for (int t = wave; t < tiles; t += WAVES) {
    const int tm = (t / nT) << 4;
    const int tn = (t % nT) << 4;
    const int m = tm + l16;
    const int n = tn + l16;
    v8f acc = {};
    for (int k0 = 0; k0 < K; k0 += 4) {
      const int ka = k0 + (half << 1);  // A/B fragment: K = 2*(lane/16) + v
      v2f av, bv;
      if constexpr (!TA) {
        av = *(const v2f*)(A + m * lda + ka);      // contiguous pair -> b64
      } else {
        av[0] = A[(ka + 0) * lda + m];
        av[1] = A[(ka + 1) * lda + m];
      }
      if constexpr (TB) {
        bv = *(const v2f*)(B + n * ldb + ka);      // contiguous pair -> b64
      } else {
        bv[0] = B[(ka + 0) * ldb + n];
        bv[1] = B[(ka + 1) * ldb + n];
      }
#ifdef HAVE_WMMA_F32X4
      acc = __builtin_amdgcn_wmma_f32_16x16x4_f32(
          /*neg_a=*/false, av, /*neg_b=*/false, bv,
          /*c_mod=*/(short)0, acc, /*reuse_a=*/false, /*reuse_b=*/false);
#else
      acc[0] += av[0] * bv[0] + av[1] * bv[1];  // placeholder fallback
#endif
    }
    // C/D fragment layout: row = tm + v + 8*(lane/16), col = tn + lane%16
#pragma unroll
    for (int v = 0; v < 8; ++v) {
      const int row = tm + v + (half << 3);
      C[row * ldc + tn + l16] = acc[v];
    }
  }
}

}  // namespace

__global__ __launch_bounds__(256)
void peps_amplitude_kernel(const float* __restrict__ T,
                           const int* __restrict__ x,
                           float* __restrict__ out,
                           float* __restrict__ ws) {
  const int blk = blockIdx.x;  // one config per block
  float* wsb = ws + (size_t)blk * WS_PER;
  float* AS = wsb + OFF_AS;   // [i][y][u][d][l][r], 4^4 per site
  float* Lm = wsb + OFF_L;    // left MPS  [i][a][p][b] (32,4,32)
  float* Rm = wsb + OFF_R;    // right MPS
  float* X  = wsb + OFF_X;    // expanded sites [i][l=128][p=4][r=128]
  float* SC = wsb + OFF_SC;   // scratch
  float* G  = wsb + OFF_G;    // Gram

  __shared__ float sM1[BDX * BD];  // 4096: absorb operand (a,b) x p
  __shared__ float sA2[256];       // p x (c,d,newphys)
  __shared__ float sV[BDX * BD];   // 4096: subspace basis 128x32
  __shared__ float sT1[BD * BD];   // 1024: V^T V
  __shared__ float sT2[BDX * BD];  // 4096: V*T1 / G*V / final W (32x128 view)
  __shared__ float sE[BD * BD];    // 1024: environment
  __shared__ float sF[BD * BD];    // 1024

  // ---- gather A[i,y] = T[i,y,cfg[i,y]] ----
  for (int e = threadIdx.x; e < AS_SZ; e += NT) {
    const int i = e / (LY * 256);
    const int y = (e / 256) % LY;
    const int k = e & 255;
    const int cfg = x[blk * (LX * LY) + i * LY + y];
    const int src = (((i * LY + y) * PHYS + cfg) << 8) + k;
    __builtin_prefetch(T + src + NT, 0, 1);  // global_prefetch_b8 next tile
    AS[e] = T[src];
  }
  __syncthreads();

  // ---- two boundary sweeps: side 0 = left, side 1 = right ----
  for (int side = 0; side < 2; ++side) {
    float* Mp = side ? Rm : Lm;
    zero_f(Mp, MPS_SZ);
    __syncthreads();

    // init boundary MPS from edge column
    const int ycol0 = side ? (LY - 1) : 0;
    for (int i = 0; i < LX; ++i) {
      const float* A0 = AS + (i * LY + ycol0) * 256;
      float* Ms = Mp + i * (BD * D * BD);
      const int amax = (i == 0) ? 1 : D;
      const int bmax = (i == LX - 1) ? 1 : D;
      for (int e = threadIdx.x; e < 64; e += NT) {
        const int a = e >> 4, p = (e >> 2) & 3, b = e & 3;
        if (a < amax && b < bmax) {
          // left:  M[a,p,b] = A[a,b,l=0,p]; right: M[a,p,b] = A[a,b,p,r=0]
          const float v = side ? A0[((a * 4 + b) * 4 + p) * 4 + 0]
                               : A0[((a * 4 + b) * 4 + 0) * 4 + p];
          Ms[(a * D + p) * BD + b] = v;
        }
      }
    }
    __syncthreads();

    // absorb columns toward the middle
    for (int yy = 1; yy < LY / 2; ++yy) {
      const int y = side ? (LY - 1 - yy) : yy;
      // --- absorb column y into every site: X[i] = einsum('apb,cd?r->acrbd')
      for (int i = 0; i < LX; ++i) {
        const float* Ac = AS + (i * LY + y) * 256;
        const float* Ms = Mp + i * (BD * D * BD);
        float* Xi = X + i * XS_SZ;
        // sA2[p][(c,d,q)]: q = new phys (right virt for left sweep, left for right)
        for (int e = threadIdx.x; e < 256; e += NT) {
          const int p = e >> 6, c = (e >> 4) & 3, d = (e >> 2) & 3, q = e & 3;
          float v = side ? Ac[((c * 4 + d) * 4 + q) * 4 + p]
                         : Ac[((c * 4 + d) * 4 + p) * 4 + q];
          if (i == 0 && c) v = 0.f;
          if (i == LX - 1 && d) v = 0.f;
          sA2[e] = v;
        }
        // sM1[(a*32+b)][p] = M[a,p,b]
        for (int e = threadIdx.x; e < BDX * BD; e += NT) {
          const int row = e >> 2, p = e & 3;
          const int a = row >> 5, b = row & 31;
          sM1[e] = Ms[(a * D + p) * BD + b];
        }
        __syncthreads();
        // SC(1024x64) = sM1(1024x4) * sA2(4x64)  -- K=4: one WMMA per tile
        gemm<false, false>(sM1, 4, sA2, 64, SC, 64, 1024, 64, 4);
        __syncthreads();
        // scatter (a,b)x(c,d,q) -> Xi[(a*4+c)][q][(b*4+d)]  (full overwrite)
        for (int e = threadIdx.x; e < 1024 * 64; e += NT) {
          const int row = e >> 6, col = e & 63;
          const int a = row >> 5, b = row & 31;
          const int c = col >> 4, d = (col >> 2) & 3, q = col & 3;
          Xi[(size_t)(a * 4 + c) * (D * BDX) + q * BDX + (b * 4 + d)] = SC[e];
        }
        __syncthreads();
      }

      // --- truncation sweep bottom->top: keep top-32 row space per bond ---
      for (int i = LX - 1; i >= 1; --i) {
        float* Xi = X + i * XS_SZ;
        float* Xp = X + (i - 1) * XS_SZ;
        // Gram G(128x128) = M M^T, M = Xi viewed (128 x 512)
        gemm<false, true>(Xi, D * BDX, Xi, D * BDX, G, BDX, BDX, BDX, D * BDX);
        __syncthreads();
        // V = G[:, 0:32] as subspace seed
        for (int e = threadIdx.x; e < BDX * BD; e += NT)
          sV[e] = G[(e >> 5) * BDX + (e & 31)];
        __syncthreads();
        // power + Newton-Schulz orthogonalization (matmuls only)
        for (int it = 0; it < 3; ++it) {
          gemm<false, false>(G, BDX, sV, BD, sT2, BD, BDX, BD, BDX);  // T2=G*V
          __syncthreads();
          copy_f(sV, sT2, BDX * BD);
          __syncthreads();
          gemm<true, false>(sV, BD, sV, BD, sT1, BD, BD, BD, BDX);    // T1=V^T V
          __syncthreads();
          gemm<false, false>(sV, BD, sT1, BD, sT2, BD, BDX, BD, BD);  // T2=V*T1
          __syncthreads();
          for (int e = threadIdx.x; e < BDX * BD; e += NT)
            sV[e] = 1.5f * sV[e] - 0.5f * sT2[e];
          __syncthreads();
        }
        // new site: SC(32x512) = V^T * M   (== S V^T of the SVD row space)
        gemm<true, false>(sV, BD, Xi, D * BDX, SC, D * BDX, BD, D * BDX, BDX);
        __syncthreads();
        copy_f(Xi, SC, BD * D * BDX);
        zero_f(Xi + BD * D * BDX, (BDX - BD) * D * BDX);
        __syncthreads();
        // neighbor: Xp (512x128) * V (128x32) -> right bond truncated to 32
        gemm<false, false>(Xp, BDX, sV, BD, SC, BD, BDX * D, BD, BDX);
        __syncthreads();
        for (int e = threadIdx.x; e < BDX * D * BDX; e += NT) {
          const int row = e >> 7, c = e & 127;
          Xp[e] = (c < BD) ? SC[row * BD + c] : 0.f;
        }
        __syncthreads();
      }
      // compact (32,4,32) blocks back into the boundary MPS
      for (int i = 0; i < LX; ++i) {
        const float* Xi = X + i * XS_SZ;
        float* Ms = Mp + i * (BD * D * BD);
        for (int e = threadIdx.x; e < BD * D * BD; e += NT) {
          const int a = e >> 7, p = (e >> 5) & 3, b = e & 31;
          Ms[e] = Xi[a * (D * BDX) + p * BDX + b];
        }
      }
      __syncthreads();
    }
  }

  // ---- zip-up: E <- einsum('ab,apc,bpd->cd', E, L[i], R[i]) ----
  for (int e = threadIdx.x; e < BD * BD; e += NT) sE[e] = (e == 0) ? 1.f : 0.f;
  __syncthreads();
  for (int i = 0; i < LX; ++i) {
    const float* Lmat = Lm + i * (BD * D * BD);  // rows a, cols (p,c), ld=128
    const float* Rmat = Rm + i * (BD * D * BD);
    // W(32x128) = E^T * Lmat   (W lives in sT2)
    gemm<true, false>(sE, BD, Lmat, D * BD, sT2, D * BD, BD, D * BD, BD);
    __syncthreads();
    // F(32x32) = W~(128x32)^T * R~(128x32)  (row-major reinterpretation)
    gemm<true, false>(sT2, BD, Rmat, BD, sF, BD, BD, BD, D * BD);
    __syncthreads();
    copy_f(sE, sF, BD * BD);
    __syncthreads();
  }
  if (threadIdx.x == 0) out[blk] = sE[0];
}

extern "C" void kernel_launch(void* const* d_in, const int* in_sizes, int n_in,
                              void* d_out, int out_size, void* d_ws,
                              size_t ws_size, hipStream_t stream) {
  (void)in_sizes; (void)n_in; (void)out_size; (void)ws_size;
  const float* T = (const float*)d_in[0];  // (10,10,2,4,4,4,4) f32
  const int* x = (const int*)d_in[1];      // (64,100) int32
  float* out = (float*)d_out;              // (64,) f32
  float* ws = (float*)d_ws;
  peps_amplitude_kernel<<<dim3(BATCH), dim3(NT), 0, stream>>>(T, x, out, ws);
}